// PosGateModel_56109452755471
// MI455X (gfx1250) — compile-verified
//
#include <hip/hip_runtime.h>

typedef __attribute__((ext_vector_type(2))) float v2f;
typedef __attribute__((ext_vector_type(8))) float v8f;

#define HDIM 128
#define H2DIM 256
#define VDIM 32000
#define BATCH 64
#define SEQL 2048

// D(16x16,f32) += A(16x4,f32) * B(4x16,f32)  — native CDNA5 f32 WMMA
__device__ __forceinline__ v8f wmma_f32_k4(v8f acc, v2f a, v2f b) {
  return __builtin_amdgcn_wmma_f32_16x16x4_f32(false, a, false, b, (short)0, acc,
                                               false, false);
}

// Async DMA: 16 bytes/lane, global -> LDS, ASYNCcnt-tracked (CDNA5).
// GV form: per-lane 64-bit global address in a VGPR pair; LDS byte offset is
// the low 32 bits of the generic pointer (LDS aperture stores offset there).
__device__ __forceinline__ void async_g2l_b128(const void* gaddr, void* lds) {
  unsigned ldst = (unsigned)(uintptr_t)lds;
  asm volatile("global_load_async_to_lds_b128 %0, %1, off"
               :
               : "v"(ldst), "v"(gaddr)
               : "memory");
}
__device__ __forceinline__ void wait_async0() {
  asm volatile("s_wait_asynccnt 0x0" ::: "memory");
}

// ---------------------------------------------------------------------------
// Kernel 1: fused embed-gather + MLP(relu) + residual + LayerNorm + key proj.
// 64 threads = 2 waves; each wave owns a 16-token tile. Embedding rows are
// gathered by async DMA straight into LDS; all GEMMs via f32 WMMA.
// ---------------------------------------------------------------------------
__global__ __launch_bounds__(64) void fused_embed_mlp_ln_keys(
    const int* __restrict__ seq, const float* __restrict__ embed,
    const float* __restrict__ w1, const float* __restrict__ b1,
    const float* __restrict__ w2, const float* __restrict__ b2,
    const float* __restrict__ ln_g, const float* __restrict__ ln_b,
    const float* __restrict__ kp_w, float* __restrict__ keys) {
  __shared__ __align__(16) float s_x[2][16][HDIM];    // h, then x/hn in place
  __shared__ __align__(16) float s_y1[2][16][H2DIM];  // relu(h@w1+b1)

  const int wave = threadIdx.x >> 5;
  const int lane = threadIdx.x & 31;
  const int m = lane & 15;   // A row / B,C,D column within tile
  const int kh = lane >> 4;  // selects K pair {0,1} vs {2,3}
  const int tok0 = (blockIdx.x * 2 + wave) * 16;

  // Async-gather 16 embedding rows (512 B each: 4 floats/lane) into LDS.
  for (int r = 0; r < 16; ++r) {
    const float* rowp = embed + (size_t)seq[tok0 + r] * HDIM + lane * 4;
    async_g2l_b128(rowp, &s_x[wave][r][lane * 4]);
  }
  wait_async0();
  __syncthreads();

  // GEMM1: y1 = relu(h @ w1 + b1), 16x256, K=128.
  for (int nt = 0; nt < H2DIM / 16; ++nt) {
    v8f acc = {};
    for (int kk = 0; kk < HDIM; kk += 4) {
      const int k0 = kk + 2 * kh;
      v2f a;  a.x = s_x[wave][m][k0];
              a.y = s_x[wave][m][k0 + 1];
      v2f bb; bb.x = w1[k0 * H2DIM + nt * 16 + m];
              bb.y = w1[(k0 + 1) * H2DIM + nt * 16 + m];
      acc = wmma_f32_k4(acc, a, bb);
    }
    const int col = nt * 16 + m;
    const float bias = b1[col];
    for (int j = 0; j < 8; ++j) {
      const float v = acc[j] + bias;
      s_y1[wave][j + 8 * kh][col] = v > 0.0f ? v : 0.0f;
    }
  }
  __syncthreads();

  // GEMM2: x = h + y1 @ w2 + b2 (in place over s_x), 16x128, K=256.
  for (int nt = 0; nt < HDIM / 16; ++nt) {
    v8f acc = {};
    for (int kk = 0; kk < H2DIM; kk += 4) {
      const int k0 = kk + 2 * kh;
      v2f a;  a.x = s_y1[wave][m][k0];
              a.y = s_y1[wave][m][k0 + 1];
      v2f bb; bb.x = w2[k0 * HDIM + nt * 16 + m];
              bb.y = w2[(k0 + 1) * HDIM + nt * 16 + m];
      acc = wmma_f32_k4(acc, a, bb);
    }
    const int col = nt * 16 + m;
    const float bias = b2[col];
    for (int j = 0; j < 8; ++j)
      s_x[wave][j + 8 * kh][col] += acc[j] + bias;  // residual, unique owner
  }
  __syncthreads();

  // LayerNorm: lanes 0..15 each normalize one row.
  if (lane < 16) {
    float mu = 0.0f;
    for (int c = 0; c < HDIM; ++c) mu += s_x[wave][lane][c];
    mu *= (1.0f / HDIM);
    float var = 0.0f;
    for (int c = 0; c < HDIM; ++c) {
      const float d = s_x[wave][lane][c] - mu;
      var += d * d;
    }
    var *= (1.0f / HDIM);
    const float inv = rsqrtf(var + 1e-5f);
    for (int c = 0; c < HDIM; ++c)
      s_x[wave][lane][c] = (s_x[wave][lane][c] - mu) * inv * ln_g[c] + ln_b[c];
  }
  __syncthreads();  // reconverged: EXEC all-1s before WMMA

  // GEMM3: keys = hn @ kp_w, 16x128, K=128.
  for (int nt = 0; nt < HDIM / 16; ++nt) {
    v8f acc = {};
    for (int kk = 0; kk < HDIM; kk += 4) {
      const int k0 = kk + 2 * kh;
      v2f a;  a.x = s_x[wave][m][k0];
              a.y = s_x[wave][m][k0 + 1];
      v2f bb; bb.x = kp_w[k0 * HDIM + nt * 16 + m];
              bb.y = kp_w[(k0 + 1) * HDIM + nt * 16 + m];
      acc = wmma_f32_k4(acc, a, bb);
    }
    const int col = nt * 16 + m;
    for (int j = 0; j < 8; ++j)
      keys[(size_t)(tok0 + j + 8 * kh) * HDIM + col] = acc[j];
  }
}

// ---------------------------------------------------------------------------
// Kernel 2: sequential fast-weight scan. One workgroup per batch element,
// M (128x128 f32 = 64 KiB) resident in LDS. 256 threads: 2 per row of M.
// Key rows double-buffered in LDS; row t+1 prefetched by async DMA while
// step t computes, hiding all global latency on the serial critical path.
// ---------------------------------------------------------------------------
__global__ __launch_bounds__(256) void memory_scan(
    const float* __restrict__ keys, float* __restrict__ readv) {
  __shared__ float M[HDIM][HDIM];               // 64 KiB
  __shared__ __align__(16) float kstage[2][HDIM];  // double-buffered key rows
  __shared__ float knbuf[HDIM];
  __shared__ float errbuf[HDIM];
  __shared__ float red[HDIM];
  __shared__ float scal[2];  // [0] = ||k||^2, [1] = ||err||^2

  const int b = blockIdx.x;
  const int tid = threadIdx.x;
  const int row = tid >> 1;
  const int h0 = (tid & 1) * 64;
  const float* kbase = keys + (size_t)b * SEQL * HDIM;

  for (int i = tid; i < HDIM * HDIM; i += 256) (&M[0][0])[i] = 0.0f;

  // Prologue: async-stage key row 0.
  if (tid < 32) async_g2l_b128(kbase + tid * 4, &kstage[0][tid * 4]);
  wait_async0();
  __syncthreads();

  const float tscale = 1.0f / (float)(SEQL - 1);
  for (int t = 0; t < SEQL - 1; ++t) {
    const int cur = t & 1;
    const float* ks = kstage[cur];

    // Prefetch key row t+1 into the other buffer (overlaps with compute).
    if (tid < 32)
      async_g2l_b128(kbase + (size_t)(t + 1) * HDIM + tid * 4,
                     &kstage[cur ^ 1][tid * 4]);

    if (tid < HDIM) red[tid] = ks[tid] * ks[tid];
    __syncthreads();
    if (tid < 32) {
      float v = red[tid] + red[tid + 32] + red[tid + 64] + red[tid + 96];
      v += __shfl_xor(v, 16); v += __shfl_xor(v, 8); v += __shfl_xor(v, 4);
      v += __shfl_xor(v, 2);  v += __shfl_xor(v, 1);
      if (tid == 0) scal[0] = v;
    }
    __syncthreads();
    const float knorm2 = scal[0];
    const float inv = 1.0f / fmaxf(sqrtf(knorm2), 1e-12f);
    if (tid < HDIM) knbuf[tid] = ks[tid] * inv;
    __syncthreads();

    // vp = M @ kn ; err = k - vp
    float p = 0.0f;
#pragma unroll
    for (int j = 0; j < 64; ++j) p += M[row][h0 + j] * knbuf[h0 + j];
    p += __shfl_xor(p, 1);  // combine the two half-row partials
    const float e = ks[row] - p;
    if ((tid & 1) == 0) { errbuf[row] = e; red[row] = e * e; }
    __syncthreads();
    if (tid < 32) {
      float v = red[tid] + red[tid + 32] + red[tid + 64] + red[tid + 96];
      v += __shfl_xor(v, 16); v += __shfl_xor(v, 8); v += __shfl_xor(v, 4);
      v += __shfl_xor(v, 2);  v += __shfl_xor(v, 1);
      if (tid == 0) scal[1] = v;
    }
    __syncthreads();

    const float th = 0.4f + 1.1f * (1.0f - (float)t * tscale);
    if (scal[1] >= th * th * knorm2) {  // uniform across block
      const float ce = 0.05f * errbuf[row];  // (1 - ALPHA) * err[i]
#pragma unroll
      for (int j = 0; j < 64; ++j) M[row][h0 + j] += ce * knbuf[h0 + j];
    }
    wait_async0();   // row t+1 landed in LDS
    __syncthreads(); // make it (and M update) visible block-wide
  }

  // read = M @ q, q = keys[b, L-1, :] (already staged in kstage[1]).
  const float* q = kstage[1];
  float p = 0.0f;
#pragma unroll
  for (int j = 0; j < 64; ++j) p += M[row][h0 + j] * q[h0 + j];
  p += __shfl_xor(p, 1);
  if ((tid & 1) == 0) readv[b * HDIM + row] = p;
}

// ---------------------------------------------------------------------------
// Kernel 3a: tmp = read @ rp_w + rp_b  (64x128 @ 128x128), one block, WMMA.
// ---------------------------------------------------------------------------
__global__ __launch_bounds__(256) void rp_gemm(
    const float* __restrict__ readv, const float* __restrict__ rp_w,
    const float* __restrict__ rp_b, float* __restrict__ tmp) {
  __shared__ __align__(16) float sA[BATCH * HDIM];  // 32 KiB
  const int tid = threadIdx.x;
  for (int r = 0; r < BATCH * HDIM; r += 1024) {
    const int idx = r + tid * 4;
    async_g2l_b128(readv + idx, &sA[idx]);
  }
  wait_async0();
  __syncthreads();
  const int wave = tid >> 5, lane = tid & 31;
  const int m = lane & 15, kh = lane >> 4;
  for (int tile = wave; tile < 32; tile += 8) {  // 4 M-tiles x 8 N-tiles
    const int mt = tile & 3, nt = tile >> 2;
    v8f acc = {};
    for (int kk = 0; kk < HDIM; kk += 4) {
      const int k0 = kk + 2 * kh;
      v2f a;  a.x = sA[(mt * 16 + m) * HDIM + k0];
              a.y = sA[(mt * 16 + m) * HDIM + k0 + 1];
      v2f bb; bb.x = rp_w[k0 * HDIM + nt * 16 + m];
              bb.y = rp_w[(k0 + 1) * HDIM + nt * 16 + m];
      acc = wmma_f32_k4(acc, a, bb);
    }
    const int col = nt * 16 + m;
    const float bias = rp_b[col];
    for (int j = 0; j < 8; ++j)
      tmp[(mt * 16 + j + 8 * kh) * HDIM + col] = acc[j] + bias;
  }
}

// ---------------------------------------------------------------------------
// Kernel 3b: out = tmp @ out_w + out_b  (64x128 @ 128x32000), WMMA,
// streaming the 16.4 MB out_w panel (bandwidth-bound) with prefetch.
// ---------------------------------------------------------------------------
__global__ __launch_bounds__(256) void out_gemm(
    const float* __restrict__ tmp, const float* __restrict__ out_w,
    const float* __restrict__ out_b, float* __restrict__ out) {
  __shared__ __align__(16) float sA[BATCH * HDIM];  // 32 KiB
  const int tid = threadIdx.x;
  for (int r = 0; r < BATCH * HDIM; r += 1024) {
    const int idx = r + tid * 4;
    async_g2l_b128(tmp + idx, &sA[idx]);
  }
  wait_async0();
  __syncthreads();
  const int wave = tid >> 5, lane = tid & 31;
  const int m = lane & 15, kh = lane >> 4;
  const int tile = blockIdx.x * 8 + wave;  // 8000 tiles total
  const int mt = tile & 3;                 // 4 M-tiles
  const int nt = tile >> 2;                // 2000 N-tiles
  v8f acc = {};
  for (int kk = 0; kk < HDIM; kk += 4) {
    if (kk + 8 < HDIM)  // uniform: prefetch B panel two k-steps ahead
      __builtin_prefetch(&out_w[(size_t)(kk + 8) * VDIM + nt * 16], 0, 0);
    const int k0 = kk + 2 * kh;
    v2f a;  a.x = sA[(mt * 16 + m) * HDIM + k0];
            a.y = sA[(mt * 16 + m) * HDIM + k0 + 1];
    v2f bb; bb.x = out_w[(size_t)k0 * VDIM + nt * 16 + m];
            bb.y = out_w[(size_t)(k0 + 1) * VDIM + nt * 16 + m];
    acc = wmma_f32_k4(acc, a, bb);
  }
  const int col = nt * 16 + m;
  const float bias = out_b[col];
  for (int j = 0; j < 8; ++j)
    out[(size_t)(mt * 16 + j + 8 * kh) * VDIM + col] = acc[j] + bias;
}

// ---------------------------------------------------------------------------
extern "C" void kernel_launch(void* const* d_in, const int* in_sizes, int n_in,
                              void* d_out, int out_size, void* d_ws,
                              size_t ws_size, hipStream_t stream) {
  (void)in_sizes; (void)n_in; (void)out_size; (void)ws_size;
  const int*   seq   = (const int*)  d_in[0];
  const float* embed = (const float*)d_in[1];
  const float* w1    = (const float*)d_in[2];
  const float* b1    = (const float*)d_in[3];
  const float* w2    = (const float*)d_in[4];
  const float* b2    = (const float*)d_in[5];
  const float* ln_g  = (const float*)d_in[6];
  const float* ln_b  = (const float*)d_in[7];
  const float* kp_w  = (const float*)d_in[8];
  const float* rp_w  = (const float*)d_in[9];
  const float* rp_b  = (const float*)d_in[10];
  const float* out_w = (const float*)d_in[11];
  const float* out_b = (const float*)d_in[12];
  float* out = (float*)d_out;

  float* ws   = (float*)d_ws;
  float* keys = ws;                                   // B*L*H  (64 MiB)
  float* rd   = ws + (size_t)BATCH * SEQL * HDIM;     // B*H
  float* tmp  = rd + BATCH * HDIM;                    // B*H

  fused_embed_mlp_ln_keys<<<(BATCH * SEQL) / 32, 64, 0, stream>>>(
      seq, embed, w1, b1, w2, b2, ln_g, ln_b, kp_w, keys);
  memory_scan<<<BATCH, 256, 0, stream>>>(keys, rd);
  rp_gemm<<<1, 256, 0, stream>>>(rd, rp_w, rp_b, tmp);
  out_gemm<<<(4 * (VDIM / 16)) / 8, 256, 0, stream>>>(tmp, out_w, out_b, out);
}